// LightRNNDecoder_32813550141544
// MI455X (gfx1250) — compile-verified
//
#include <hip/hip_runtime.h>

typedef float v2f __attribute__((ext_vector_type(2)));
typedef float v8f __attribute__((ext_vector_type(8)));

#define D_DIM 1024
#define R_DIM 224
#define N_TOK 4096
#define WAVES 7
#define NT_PER_WAVE 2            // 7 waves * 2 N-tiles * 16 cols = 224 = R_DIM
#define BLOCK_T (WAVES * 32)
#define LDS_STRIDE 228           // padded row stride for logits in LDS

// ---------------------------------------------------------------------------
// Bucketing: histogram of row ids -> prefix scan -> scatter permutation
// ---------------------------------------------------------------------------
__global__ void init_kernel(int* counts, int* cursors, float* sums) {
    int i = blockIdx.x * blockDim.x + threadIdx.x;
    if (i < R_DIM) { counts[i] = 0; cursors[i] = 0; }
    if (i < 2) sums[i] = 0.0f;
}

__global__ void count_kernel(const int* __restrict__ row_ids, int* counts) {
    int i = blockIdx.x * blockDim.x + threadIdx.x;
    if (i < N_TOK) atomicAdd(&counts[row_ids[i]], 1);
}

__global__ void scan_kernel(const int* __restrict__ counts, int* offsets) {
    if (threadIdx.x == 0 && blockIdx.x == 0) {
        int acc = 0;
        for (int i = 0; i < R_DIM; ++i) { offsets[i] = acc; acc += counts[i]; }
        offsets[R_DIM] = acc;
    }
}

__global__ void scatter_kernel(const int* __restrict__ row_ids,
                               const int* __restrict__ offsets,
                               int* cursors, int* perm) {
    int i = blockIdx.x * blockDim.x + threadIdx.x;
    if (i < N_TOK) {
        int r = row_ids[i];
        int pos = offsets[r] + atomicAdd(&cursors[r], 1);
        perm[pos] = i;
    }
}

// ---------------------------------------------------------------------------
// Row loss: logits = hs @ w_row + b_row, per-token CE, sum into sums[0].
// One block = one 16-token tile. 7 waves, each owning 2 column tiles of 16.
// A fragment (16x4 f32): lane&15 = M, (lane>>4)*2 = K offset within step.
// ---------------------------------------------------------------------------
__global__ __launch_bounds__(BLOCK_T)
void row_loss_kernel(const float* __restrict__ hs,
                     const float* __restrict__ w_row,
                     const float* __restrict__ b_row,
                     const int*  __restrict__ row_ids,
                     float* __restrict__ sums) {
    __shared__ float lds[16 * LDS_STRIDE];

    const int wave = threadIdx.x >> 5;
    const int lane = threadIdx.x & 31;
    const int m0   = blockIdx.x * 16;          // token tile base
    const int Ma   = lane & 15;                // A-fragment row / B,C column lane
    const int koff = (lane >> 4) * 2;          // K offset within a K=4 step
    const int n_base = wave * (NT_PER_WAVE * 16);

    const float* arow  = hs + (size_t)(m0 + Ma) * D_DIM + koff;
    const float* bcol0 = w_row + n_base + Ma;

    v8f acc[NT_PER_WAVE] = {};
    for (int k = 0; k < D_DIM; k += 4) {
        v2f a;
        a.x = arow[k];
        a.y = arow[k + 1];
#pragma unroll
        for (int t = 0; t < NT_PER_WAVE; ++t) {
            const float* bp = bcol0 + (size_t)(k + koff) * R_DIM + t * 16;
            v2f b;
            b.x = bp[0];
            b.y = bp[R_DIM];
            acc[t] = __builtin_amdgcn_wmma_f32_16x16x4_f32(
                false, a, false, b, (short)0, acc[t], false, false);
        }
    }

    // C layout: lane&15 = N, M = (lane>>4)*8 + j
#pragma unroll
    for (int t = 0; t < NT_PER_WAVE; ++t) {
        int n = n_base + t * 16 + Ma;
        float bias = b_row[n];
#pragma unroll
        for (int j = 0; j < 8; ++j) {
            int M = ((lane >> 4) << 3) + j;
            lds[M * LDS_STRIDE + n] = acc[t][j] + bias;
        }
    }
    __syncthreads();

    if (threadIdx.x < 16) {
        int tok = m0 + threadIdx.x;
        const float* lg = &lds[threadIdx.x * LDS_STRIDE];
        float mx = -3.4e38f;
        for (int n = 0; n < R_DIM; ++n) mx = fmaxf(mx, lg[n]);
        float s = 0.0f;
        for (int n = 0; n < R_DIM; ++n) s += __expf(lg[n] - mx);
        float lse = mx + __logf(s);
        float nll = lse - lg[row_ids[tok]];
        atomicAdd(&sums[0], nll);
    }
}

// ---------------------------------------------------------------------------
// Col loss: one block per row id r. Tokens of that bucket in 16-token tiles.
// logits = hs[tok] @ col_weight[r] + col_bias[r]; CE vs target_col_ids.
// ---------------------------------------------------------------------------
__global__ __launch_bounds__(BLOCK_T)
void col_loss_kernel(const float* __restrict__ hs,
                     const float* __restrict__ col_weight,
                     const float* __restrict__ col_bias,
                     const int*  __restrict__ col_ids,
                     const int*  __restrict__ offsets,
                     const int*  __restrict__ perm,
                     float* __restrict__ sums) {
    const int r   = blockIdx.x;
    const int beg = offsets[r];
    const int cnt = offsets[r + 1] - beg;
    if (cnt <= 0) return;

    const float* W    = col_weight + (size_t)r * D_DIM * R_DIM;
    const float* bias = col_bias   + (size_t)r * R_DIM;

    __shared__ float lds[16 * LDS_STRIDE];
    __shared__ int   toks[16];

    const int wave = threadIdx.x >> 5;
    const int lane = threadIdx.x & 31;
    const int Ma   = lane & 15;
    const int koff = (lane >> 4) * 2;
    const int n_base = wave * (NT_PER_WAVE * 16);

    for (int tile = 0; tile * 16 < cnt; ++tile) {
        if (threadIdx.x < 16) {
            int idx = tile * 16 + threadIdx.x;
            toks[threadIdx.x] = (idx < cnt) ? perm[beg + idx] : perm[beg];
        }
        __syncthreads();

        const float* arow  = hs + (size_t)toks[Ma] * D_DIM + koff;
        const float* bcol0 = W + n_base + Ma;

        v8f acc[NT_PER_WAVE] = {};
        for (int k = 0; k < D_DIM; k += 4) {
            v2f a;
            a.x = arow[k];
            a.y = arow[k + 1];
#pragma unroll
            for (int t = 0; t < NT_PER_WAVE; ++t) {
                const float* bp = bcol0 + (size_t)(k + koff) * R_DIM + t * 16;
                v2f b;
                b.x = bp[0];
                b.y = bp[R_DIM];
                acc[t] = __builtin_amdgcn_wmma_f32_16x16x4_f32(
                    false, a, false, b, (short)0, acc[t], false, false);
            }
        }

#pragma unroll
        for (int t = 0; t < NT_PER_WAVE; ++t) {
            int n = n_base + t * 16 + Ma;
            float bv = bias[n];
#pragma unroll
            for (int j = 0; j < 8; ++j) {
                int M = ((lane >> 4) << 3) + j;
                lds[M * LDS_STRIDE + n] = acc[t][j] + bv;
            }
        }
        __syncthreads();

        if (threadIdx.x < 16) {
            int idx = tile * 16 + threadIdx.x;
            if (idx < cnt) {
                int tok = toks[threadIdx.x];
                const float* lg = &lds[threadIdx.x * LDS_STRIDE];
                float mx = -3.4e38f;
                for (int n = 0; n < R_DIM; ++n) mx = fmaxf(mx, lg[n]);
                float s = 0.0f;
                for (int n = 0; n < R_DIM; ++n) s += __expf(lg[n] - mx);
                float lse = mx + __logf(s);
                float nll = lse - lg[col_ids[tok]];
                atomicAdd(&sums[1], nll);
            }
        }
        __syncthreads();   // protect lds/toks before next tile
    }
}

__global__ void finalize_kernel(const float* __restrict__ sums, float* out) {
    if (threadIdx.x == 0 && blockIdx.x == 0)
        out[0] = (sums[0] + sums[1]) * (1.0f / (float)N_TOK);
}

// ---------------------------------------------------------------------------
extern "C" void kernel_launch(void* const* d_in, const int* in_sizes, int n_in,
                              void* d_out, int out_size, void* d_ws, size_t ws_size,
                              hipStream_t stream) {
    (void)in_sizes; (void)n_in; (void)out_size; (void)ws_size;
    const float* hs    = (const float*)d_in[0];
    const float* w_row = (const float*)d_in[1];
    const float* b_row = (const float*)d_in[2];
    const float* col_w = (const float*)d_in[3];
    const float* col_b = (const float*)d_in[4];
    const int*   rid   = (const int*)d_in[5];
    const int*   cid   = (const int*)d_in[6];

    int*   counts  = (int*)d_ws;
    int*   cursors = counts + 256;
    int*   offsets = cursors + 256;
    int*   perm    = offsets + 256;
    float* sums    = (float*)(perm + N_TOK);
    float* out     = (float*)d_out;

    init_kernel   <<<1, 256, 0, stream>>>(counts, cursors, sums);
    count_kernel  <<<(N_TOK + 255) / 256, 256, 0, stream>>>(rid, counts);
    scan_kernel   <<<1, 32, 0, stream>>>(counts, offsets);
    scatter_kernel<<<(N_TOK + 255) / 256, 256, 0, stream>>>(rid, offsets, cursors, perm);

    row_loss_kernel<<<N_TOK / 16, BLOCK_T, 0, stream>>>(hs, w_row, b_row, rid, sums);
    col_loss_kernel<<<R_DIM, BLOCK_T, 0, stream>>>(hs, col_w, col_b, cid, offsets, perm, sums);

    finalize_kernel<<<1, 32, 0, stream>>>(sums, out);
}